// GINENet_2783138808354
// MI455X (gfx1250) — compile-verified
//
#include <hip/hip_runtime.h>

// ---------------------------------------------------------------------------
// GINENet inference for MI455X (gfx1250), wave32, V_WMMA_F32_16X16X4_F32.
// h (25.6MB) and aggr (25.6MB) stay L2-resident (192MB L2); edge_attr streams
// from HBM with non-temporal hints so it cannot evict them. Edge phase fuses
// bias-GEMM (WMMA, K=16) + gather + relu + atomic scatter-add. Gather/scatter
// addressing uses 32-bit element offsets so clang emits SADDR-base
// global_load/global_atomic with scale_offset + immediate (no per-access
// 64-bit VALU address math). Node phase fuses z=h+aggr, two K=128 WMMA GEMMs,
// BN, relus and atomic graph pooling.
// ---------------------------------------------------------------------------

typedef __attribute__((ext_vector_type(2))) float v2f;
typedef __attribute__((ext_vector_type(8))) float v8f;

#define N_NODES  50000
#define N_EDGES  1600000
#define N_GRAPHS 512
#define NLAYER   3
#define DD       128
#define EF       16
#define BN_EPS   1e-5f

// D = A(16x4) * B(4x16) + C.  (neg_a, A, neg_b, B, c_mod, C, reuse_a, reuse_b)
static __device__ __forceinline__ v8f wmma4(v2f a, v2f b, v8f c) {
    return __builtin_amdgcn_wmma_f32_16x16x4_f32(false, a, false, b, (short)0, c,
                                                 false, false);
}

__global__ void __launch_bounds__(256) gine_zero_kernel(float* __restrict__ p, int n) {
    int i = blockIdx.x * blockDim.x + threadIdx.x;
    if (i < n) p[i] = 0.0f;
}

// ---------------------------------------------------------------------------
// Fused edge phase: e = edge_attr@elW + elb; m = relu(h[src] + e);
// aggr[dst] += m.  One wave per 16-edge tile.
// A layout (16x4 f32): lanes 0-15 hold M=lane, K={0,1}; lanes 16-31 K={2,3}.
// C layout: lanes 0-15 VGPR r -> M=r; lanes 16-31 VGPR r -> M=8+r; N=lane&15.
// ---------------------------------------------------------------------------
__global__ void __launch_bounds__(256) gine_edge_kernel(
    const float* __restrict__ h, const float* __restrict__ edge_attr,
    const float* __restrict__ elW, const float* __restrict__ elb,
    const int* __restrict__ src, const int* __restrict__ dst,
    float* __restrict__ aggr)
{
    const int lane  = threadIdx.x & 31;
    const int wave  = threadIdx.x >> 5;
    const int tile  = blockIdx.x * 8 + wave;
    const int e0    = tile * 16;
    if (e0 >= N_EDGES) return;

    const int m     = lane & 15;   // A row / C column-within-tile
    const int khalf = lane >> 4;   // which K pair of the 4-wide step

    // A fragments, 4 K-steps covering K=16. Non-temporal: edge_attr is a pure
    // stream (no reuse) -- keep it out of the way of L2-resident h/aggr.
    v2f a[4];
    const float* ea = edge_attr + (size_t)(e0 + m) * EF;
    #pragma unroll
    for (int kc = 0; kc < 4; ++kc) {
        const int k = kc * 4 + khalf * 2;
        a[kc] = __builtin_nontemporal_load((const v2f*)(ea + k));
    }

    // Per-row 32-bit ELEMENT offsets (shuffled once per tile). 50000*128 fits
    // in 31 bits, so h[hoff + imm] / aggr[aoff + imm] select the SADDR +
    // vgpr-offset + scale_offset addressing mode: zero per-access VALU.
    const int mySrcOff = src[e0 + m] * DD;
    const int myDstOff = dst[e0 + m] * DD;
    int hoff[8], aoff[8];
    #pragma unroll
    for (int r = 0; r < 8; ++r) {
        const int row = r + khalf * 8;
        hoff[r] = __shfl(mySrcOff, row, 32) + m;
        aoff[r] = __shfl(myDstOff, row, 32) + m;
    }

    #pragma unroll
    for (int nt = 0; nt < 8; ++nt) {            // fully unrolled: nt*64B -> imm
        const int col = nt * 16 + m;
        const float bias = elb[col];
        v8f acc;
        #pragma unroll
        for (int r = 0; r < 8; ++r) acc[r] = bias;

        #pragma unroll
        for (int kc = 0; kc < 4; ++kc) {
            const int k = kc * 4 + khalf * 2;
            v2f b;
            b.x = elW[(size_t)k * DD + col];
            b.y = elW[(size_t)(k + 1) * DD + col];
            acc = wmma4(a[kc], b, acc);
        }

        // relu(h[src] + e) and scatter-add; 64B-coalesced half-wave accesses
        #pragma unroll
        for (int r = 0; r < 8; ++r) {
            float v = h[hoff[r] + nt * 16] + acc[r];
            v = v > 0.0f ? v : 0.0f;
            atomicAdd(&aggr[aoff[r] + nt * 16], v);
        }
    }
}

// ---------------------------------------------------------------------------
// Fused node MLP: z = h + aggr; y = relu(BN(z@W1+b1)); hn = relu(y@W2+b2);
// pool[batch] += hn.  4 waves/block, one 16-node tile per wave, 8KB LDS/wave.
// ---------------------------------------------------------------------------
__global__ void __launch_bounds__(128) gine_node_kernel(
    const float* __restrict__ h, const float* __restrict__ aggr,
    const float* __restrict__ W1, const float* __restrict__ b1,
    const float* __restrict__ bn_g, const float* __restrict__ bn_b,
    const float* __restrict__ bn_mean, const float* __restrict__ bn_var,
    const float* __restrict__ W2, const float* __restrict__ b2,
    const int* __restrict__ batch, float* __restrict__ pool, // pre-offset by l*D
    float* __restrict__ hout)
{
    __shared__ float zbuf[4][16 * DD];   // 32KB: z, then reused in-place for y

    const int lane  = threadIdx.x & 31;
    const int wave  = threadIdx.x >> 5;
    const int tile  = blockIdx.x * 4 + wave;
    const int n0    = tile * 16;
    const bool active = (n0 < N_NODES);
    const int m     = lane & 15;
    const int khalf = lane >> 4;

    // stage 0: z = h + aggr -> LDS (coalesced)
    if (active) {
        for (int i = lane; i < 16 * DD; i += 32) {
            const size_t gi = (size_t)n0 * DD + i;
            zbuf[wave][i] = h[gi] + aggr[gi];
        }
    }
    __syncthreads();

    // stage 1: GEMM1 (K=128 => 32 WMMA steps x 8 N-tiles) + BN + relu -> LDS
    if (active) {
        v8f acc[8];
        #pragma unroll
        for (int nt = 0; nt < 8; ++nt) {
            const float bias = b1[nt * 16 + m];
            #pragma unroll
            for (int r = 0; r < 8; ++r) acc[nt][r] = bias;
        }
        #pragma unroll 1
        for (int kc = 0; kc < 32; ++kc) {
            const int k = kc * 4 + khalf * 2;
            v2f a;
            a.x = zbuf[wave][m * DD + k];
            a.y = zbuf[wave][m * DD + k + 1];
            #pragma unroll
            for (int nt = 0; nt < 8; ++nt) {
                const int col = nt * 16 + m;
                v2f b;
                b.x = W1[(size_t)k * DD + col];
                b.y = W1[(size_t)(k + 1) * DD + col];
                acc[nt] = wmma4(a, b, acc[nt]);
            }
        }
        // BN(eval) + relu; write y back into this wave's LDS slice (all z
        // reads completed above; per-wave slice, lockstep wave32)
        #pragma unroll
        for (int nt = 0; nt < 8; ++nt) {
            const int col = nt * 16 + m;
            const float sc = bn_g[col] * rsqrtf(bn_var[col] + BN_EPS);
            const float mu = bn_mean[col];
            const float bb = bn_b[col];
            #pragma unroll
            for (int r = 0; r < 8; ++r) {
                const int row = r + khalf * 8;
                float v = (acc[nt][r] - mu) * sc + bb;
                zbuf[wave][row * DD + col] = v > 0.0f ? v : 0.0f;
            }
        }
    }
    __syncthreads();

    // stage 2: GEMM2 + relu -> h_next, plus pooled segment-sum into pool[batch]
    if (active) {
        v8f acc[8];
        #pragma unroll
        for (int nt = 0; nt < 8; ++nt) {
            const float bias = b2[nt * 16 + m];
            #pragma unroll
            for (int r = 0; r < 8; ++r) acc[nt][r] = bias;
        }
        #pragma unroll 1
        for (int kc = 0; kc < 32; ++kc) {
            const int k = kc * 4 + khalf * 2;
            v2f a;
            a.x = zbuf[wave][m * DD + k];
            a.y = zbuf[wave][m * DD + k + 1];
            #pragma unroll
            for (int nt = 0; nt < 8; ++nt) {
                const int col = nt * 16 + m;
                v2f b;
                b.x = W2[(size_t)k * DD + col];
                b.y = W2[(size_t)(k + 1) * DD + col];
                acc[nt] = wmma4(a, b, acc[nt]);
            }
        }
        // 32-bit element offsets for h_next rows and pool[batch] rows
        const int myBOff = batch[n0 + m] * (NLAYER * DD);
        int ooff[8], poff[8];
        #pragma unroll
        for (int r = 0; r < 8; ++r) {
            const int row = r + khalf * 8;
            ooff[r] = (n0 + row) * DD + m;
            poff[r] = __shfl(myBOff, row, 32) + m;
        }
        #pragma unroll
        for (int nt = 0; nt < 8; ++nt) {
            #pragma unroll
            for (int r = 0; r < 8; ++r) {
                float v = acc[nt][r];
                v = v > 0.0f ? v : 0.0f;
                hout[ooff[r] + nt * 16] = v;
                atomicAdd(&pool[poff[r] + nt * 16], v);
            }
        }
    }
}

// ---------------------------------------------------------------------------
// Head: out = relu(pools@lin1W + lin1b) @ lin2W + lin2b.  32 waves total.
// ---------------------------------------------------------------------------
__global__ void __launch_bounds__(256) gine_head_kernel(
    const float* __restrict__ pools,  // [512, 384]
    const float* __restrict__ lin1W,  // [384, 384]
    const float* __restrict__ lin1b,  // [384]
    const float* __restrict__ lin2W,  // [384]
    const float* __restrict__ lin2b,  // [1]
    float* __restrict__ out)          // [512]
{
    const int lane  = threadIdx.x & 31;
    const int wave  = threadIdx.x >> 5;
    const int tile  = blockIdx.x * 8 + wave;
    const int g0    = tile * 16;
    if (g0 >= N_GRAPHS) return;
    const int m     = lane & 15;
    const int khalf = lane >> 4;
    const int K     = NLAYER * DD; // 384

    float dot[8];
    #pragma unroll
    for (int r = 0; r < 8; ++r) dot[r] = 0.0f;

    #pragma unroll 1
    for (int nt = 0; nt < 24; ++nt) {         // 24 N-tiles of 16 -> 384 hidden
        const float bias = lin1b[nt * 16 + m];
        v8f acc;
        #pragma unroll
        for (int r = 0; r < 8; ++r) acc[r] = bias;
        #pragma unroll 1
        for (int kc = 0; kc < 96; ++kc) {     // K=384 -> 96 WMMA steps
            const int k = kc * 4 + khalf * 2;
            v2f a;
            a.x = pools[(size_t)(g0 + m) * K + k];
            a.y = pools[(size_t)(g0 + m) * K + k + 1];
            v2f b;
            b.x = lin1W[(size_t)k * K + nt * 16 + m];
            b.y = lin1W[(size_t)(k + 1) * K + nt * 16 + m];
            acc = wmma4(a, b, acc);
        }
        // fuse relu + dot with lin2W column weights
        const float w2 = lin2W[nt * 16 + m];
        #pragma unroll
        for (int r = 0; r < 8; ++r) {
            float v = acc[r] > 0.0f ? acc[r] : 0.0f;
            dot[r] += v * w2;
        }
    }
    // reduce across each 16-lane half (xor masks stay within the half)
    #pragma unroll
    for (int off = 8; off > 0; off >>= 1) {
        #pragma unroll
        for (int r = 0; r < 8; ++r) dot[r] += __shfl_xor(dot[r], off, 32);
    }
    if (m == 0) {  // lane 0 -> rows 0..7, lane 16 -> rows 8..15
        #pragma unroll
        for (int r = 0; r < 8; ++r)
            out[g0 + r + khalf * 8] = dot[r] + lin2b[0];
    }
}

// ---------------------------------------------------------------------------
extern "C" void kernel_launch(void* const* d_in, const int* in_sizes, int n_in,
                              void* d_out, int out_size, void* d_ws, size_t ws_size,
                              hipStream_t stream) {
    const float* x        = (const float*)d_in[0];
    const float* edge_attr= (const float*)d_in[1];
    const float* el_W     = (const float*)d_in[2];   // [3,16,128]
    const float* el_b     = (const float*)d_in[3];   // [3,128]
    const float* W1       = (const float*)d_in[4];   // [3,128,128]
    const float* b1       = (const float*)d_in[5];
    const float* bn_g     = (const float*)d_in[6];
    const float* bn_b     = (const float*)d_in[7];
    const float* bn_mean  = (const float*)d_in[8];
    const float* bn_var   = (const float*)d_in[9];
    const float* W2       = (const float*)d_in[10];
    const float* b2       = (const float*)d_in[11];
    const float* lin1_W   = (const float*)d_in[12];  // [384,384]
    const float* lin1_b   = (const float*)d_in[13];
    const float* lin2_W   = (const float*)d_in[14];  // [384,1]
    const float* lin2_b   = (const float*)d_in[15];
    const int*   edge_idx = (const int*)d_in[16];    // [2, E] (int32: JAX x64 off)
    const int*   batch    = (const int*)d_in[17];    // [N]
    const int* src = edge_idx;
    const int* dst = edge_idx + N_EDGES;

    // workspace layout
    float* ws    = (float*)d_ws;
    float* hbuf0 = ws;                                   // 50000*128
    float* hbuf1 = hbuf0 + (size_t)N_NODES * DD;
    float* aggr  = hbuf1 + (size_t)N_NODES * DD;
    float* pools = aggr  + (size_t)N_NODES * DD;         // 512*384
    float* out   = (float*)d_out;

    const int poolN = N_GRAPHS * NLAYER * DD;
    gine_zero_kernel<<<(poolN + 255) / 256, 256, 0, stream>>>(pools, poolN);

    const int aggrN      = N_NODES * DD;
    const int edgeBlocks = (N_EDGES / 16 + 7) / 8;       // 8 waves/block
    const int nodeBlocks = ((N_NODES + 15) / 16 + 3) / 4;

    const float* hcur = x;
    float* hnext = hbuf0;
    for (int l = 0; l < NLAYER; ++l) {
        gine_zero_kernel<<<(aggrN + 255) / 256, 256, 0, stream>>>(aggr, aggrN);
        gine_edge_kernel<<<edgeBlocks, 256, 0, stream>>>(
            hcur, edge_attr, el_W + (size_t)l * EF * DD, el_b + l * DD,
            src, dst, aggr);
        gine_node_kernel<<<nodeBlocks, 128, 0, stream>>>(
            hcur, aggr,
            W1 + (size_t)l * DD * DD, b1 + l * DD,
            bn_g + l * DD, bn_b + l * DD, bn_mean + l * DD, bn_var + l * DD,
            W2 + (size_t)l * DD * DD, b2 + l * DD,
            batch, pools + l * DD, hnext);
        hcur  = hnext;
        hnext = (l == 0) ? hbuf1 : hbuf0;
    }

    gine_head_kernel<<<(N_GRAPHS / 16 + 7) / 8, 256, 0, stream>>>(
        pools, lin1_W, lin1_b, lin2_W, lin2_b, out);
}